// PyramidROIAlign_AFP_86861418594941
// MI455X (gfx1250) — compile-verified
//
#include <hip/hip_runtime.h>
#include <hip/hip_bf16.h>
#include <stdint.h>

// PyramidROIAlign (crop_and_resize, bilinear, extrapolation_value=0) for MI455X.
// Memory-bound gather: optimized for 23.3 TB/s HBM + 192 MB L2.
//  - b128 coalesced traffic along C=256
//  - CDNA5 async global->LDS staging (ASYNCcnt) double-buffered per block
//  - nontemporal b128 stores (write-once output must not evict the ~178MB
//    feature-map working set from L2)

typedef float v4f __attribute__((ext_vector_type(4)));

#define NLVL   4
#define B_     2
#define NBOX   1000
#define C_     256
#define POOL   7
#define CELLS  49          // 7*7
#define CPR    4           // cells per round (4*64 float4 = 256 threads)
#define ROUNDS 13          // ceil(49/4)
#define LVL_OUT ((size_t)B_ * NBOX * CELLS * C_)   // 25,088,000 floats per level

__device__ __forceinline__ void async_ld_b128(uint32_t lds_byte_off, const void* gaddr) {
    // CDNA5: GLOBAL_LOAD_ASYNC_TO_LDS_B128 — per-lane 16B global->LDS, tracked by ASYNCcnt.
    asm volatile("global_load_async_to_lds_b128 %0, %1, off"
                 :: "v"(lds_byte_off), "v"(gaddr)
                 : "memory");
}

__global__ __launch_bounds__(256)
void pyramid_roialign_kernel(const float* __restrict__ boxes,
                             const float* __restrict__ fm2,
                             const float* __restrict__ fm3,
                             const float* __restrict__ fm4,
                             const float* __restrict__ fm5,
                             float* __restrict__ out)
{
    // 2 buffers x 4 cells x 4 corners x 256 floats = 32 KB (WGP has 320 KB)
    __shared__ __align__(16) float sbuf[2 * CPR * 4 * C_];
    __shared__ int   s_i0[16], s_i1[16];   // [0..6]=y rows (ti/bi), [7..13]=x cols (li/ri)
    __shared__ float s_l[16],  s_v[16];    // frac weight, validity

    const int bid   = blockIdx.x;            // 0..7999
    const int level = bid / (B_ * NBOX);
    const int rem   = bid - level * (B_ * NBOX);
    const int b     = rem / NBOX;
    const int n     = rem - b * NBOX;

    const int   H   = 256 >> level;          // square maps: 256,128,64,32
    const float Sm1 = (float)(H - 1);
    const float* fm = (level == 0) ? fm2 : (level == 1) ? fm3 : (level == 2) ? fm4 : fm5;

    const int t = threadIdx.x;

    // ---- separable interpolation table (reference: in = c1*(S-1) + i*(c2-c1)*(S-1)/6) ----
    if (t < 2 * POOL) {
        const float* bx = boxes + ((size_t)(b * NBOX + n)) * 4;   // (x1,y1,x2,y2)
        const float x1 = bx[0], y1 = bx[1], x2 = bx[2], y2 = bx[3];
        const int axis = (t >= POOL);        // 0 = y, 1 = x
        const int i    = t - axis * POOL;
        const float c1 = axis ? x1 : y1;
        const float c2 = axis ? x2 : y2;
        const float pos = fmaf((float)i, (c2 - c1) * Sm1 * (1.0f / 6.0f), c1 * Sm1);
        const float fl = floorf(pos);
        const float ce = ceilf(pos);
        s_l[t]  = pos - fl;
        s_v[t]  = (pos >= 0.0f && pos <= Sm1) ? 1.0f : 0.0f;
        s_i0[t] = (int)fminf(fmaxf(fl, 0.0f), Sm1);
        s_i1[t] = (int)fminf(fmaxf(ce, 0.0f), Sm1);
    }
    __syncthreads();

    const int lane = t & 31;
    const int wv   = t >> 5;                 // wave id 0..7

    // Stage one round (4 cells x 4 corner vectors x 1KB) via async global->LDS.
    // 32 wave-issues of 512B each; wave w does issues w*4..w*4+3 -> exactly 4
    // ASYNCcnt increments per wave per round (uniform, no divergence).
    auto stage = [&](int r, int bufsel) {
        #pragma unroll
        for (int j = 0; j < 4; ++j) {
            const int s      = wv * 4 + j;       // 0..31
            const int cir    = s >> 3;           // cell-in-round 0..3
            const int corner = (s >> 1) & 3;     // 0=tl 1=tr 2=bl 3=br
            const int half   = s & 1;            // which 128-float half of C
            int cell = r * CPR + cir;
            cell = (cell > CELLS - 1) ? (CELLS - 1) : cell;   // tail clamp (safe dup)
            const int cy = cell / POOL;
            const int cx = cell - cy * POOL;
            const int row = (corner & 2) ? s_i1[cy]        : s_i0[cy];
            const int col = (corner & 1) ? s_i1[POOL + cx] : s_i0[POOL + cx];
            const float* g = fm + ((size_t)((b * H + row) * H + col)) * C_
                                + half * 128 + lane * 4;
            const uint32_t lds_off = (uint32_t)(size_t)
                &sbuf[bufsel * (CPR * 4 * C_) + (cir * 4 + corner) * C_ + half * 128 + lane * 4];
            async_ld_b128(lds_off, (const void*)g);
        }
    };

    stage(0, 0);

    const size_t outBase = (size_t)level * LVL_OUT
                         + ((size_t)(b * NBOX + n)) * (CELLS * C_);
    const int cir = t >> 6;                  // cell-in-round this thread blends
    const int c4  = t & 63;                  // float4 index within C

    for (int r = 0; r < ROUNDS; ++r) {
        if (r + 1 < ROUNDS) {
            stage(r + 1, (r + 1) & 1);
            // 8 outstanding per wave; <=4 left ==> round r's 4 loads landed (in-order).
            asm volatile("s_wait_asynccnt 4" ::: "memory");
        } else {
            asm volatile("s_wait_asynccnt 0" ::: "memory");
        }
        __syncthreads();                     // make all waves' LDS writes visible

        const int  cell = r * CPR + cir;
        const bool live = (cell < CELLS);
        const int  cc   = live ? cell : (CELLS - 1);
        const int  cy   = cc / POOL;
        const int  cx   = cc - cy * POOL;
        const float ly  = s_l[cy];
        const float lx  = s_l[POOL + cx];
        const float msk = s_v[cy] * s_v[POOL + cx];

        const float* sb = &sbuf[(r & 1) * (CPR * 4 * C_) + cir * 4 * C_ + c4 * 4];
        const v4f tl = *(const v4f*)(sb);
        const v4f tr = *(const v4f*)(sb + C_);
        const v4f bl = *(const v4f*)(sb + 2 * C_);
        const v4f br = *(const v4f*)(sb + 3 * C_);

        v4f val;
        #pragma unroll
        for (int k = 0; k < 4; ++k) {
            const float tp = fmaf(lx, tr[k] - tl[k], tl[k]);
            const float bt = fmaf(lx, br[k] - bl[k], bl[k]);
            val[k] = fmaf(ly, bt - tp, tp) * msk;
        }

        if (live) {
            // write-once output: bypass L2 residency (nontemporal b128 store)
            __builtin_nontemporal_store(val, (v4f*)(out + outBase + (size_t)cell * C_ + c4 * 4));
        }
        __syncthreads();                     // buffer reuse fence before next stage
    }
}

extern "C" void kernel_launch(void* const* d_in, const int* in_sizes, int n_in,
                              void* d_out, int out_size, void* d_ws, size_t ws_size,
                              hipStream_t stream) {
    const float* boxes = (const float*)d_in[0];
    const float* fm2   = (const float*)d_in[1];
    const float* fm3   = (const float*)d_in[2];
    const float* fm4   = (const float*)d_in[3];
    const float* fm5   = (const float*)d_in[4];
    float* out = (float*)d_out;

    dim3 grid(NLVL * B_ * NBOX);   // 8000 blocks: one per (level, batch, box)
    dim3 block(256);               // 8 wave32s
    hipLaunchKernelGGL(pyramid_roialign_kernel, grid, block, 0, stream,
                       boxes, fm2, fm3, fm4, fm5, out);
}